// FactorizedSpectralConv2d_91259465105957
// MI455X (gfx1250) — compile-verified
//
#include <hip/hip_runtime.h>
#include <hip/hip_bf16.h>

// ---------------------------------------------------------------------------
// FactorizedSpectralConv2d for MI455X (gfx1250).
// Truncated-DFT-as-GEMM formulation, all GEMMs via V_WMMA_F32_16X16X4_F32.
// B=16, CIN=COUT=64, H=W=256, M1=M2=32.
// HBM-bound (~512 MiB traffic -> ~23 us floor); fp32 WMMA keeps reference
// precision while providing far more FLOPs than the ~28 GFLOP required.
// ---------------------------------------------------------------------------

typedef float v2f __attribute__((ext_vector_type(2)));
typedef float v8f __attribute__((ext_vector_type(8)));

__device__ __forceinline__ v8f wmma4(v2f a, v2f b, v8f c) {
  // D = A(16x4) * B(4x16) + C(16x16), fp32 throughout.
  return __builtin_amdgcn_wmma_f32_16x16x4_f32(false, a, false, b,
                                               (short)0, c, false, false);
}
__device__ __forceinline__ v8f zero8() {
  v8f z = {0.f, 0.f, 0.f, 0.f, 0.f, 0.f, 0.f, 0.f};
  return z;
}

// ---- workspace layout (float offsets) -------------------------------------
// Basis tables are pre-packed into the WMMA operand lane/VGPR layout:
//   A(16x4, 32-bit): lane l, vgpr v -> Amat[l%16][k0 + v + 2*(l/16)]
//   B(4x16, 32-bit): lane l, vgpr v -> Bmat[k0 + v + 2*(l/16)][l%16]
//   D(16x16):        lane l, vgpr r -> Dmat[8*(l/16)+r][l%16]
#define OFF_FW   0u        // fwd-W  B-packed: [ct(2)][kyt(2)][q(64)][l(32)][v(2)]  16384
#define OFF_FH   16384u    // fwd-H  A-packed: [p(3)][kxt(4)][q(64)][l(32)][v(2)]   49152
#define OFF_IH   65536u    // inv-H  A-packed: [p(3)][ht(16)][q(16)][l(32)][v(2)]   49152
#define OFF_IW   114688u   // inv-W  B-packed: [ct(2)][wt(16)][q(8)][l(32)][v(2)]   16384
#define OFF_XFT  131072u   // Xft: [b(16)][kx(64)][part(2)][i(64)][ky(32)]        4194304
#define OFF_YFT  131072u   // Yft overlays Xft (Xft dead after k_mix1)
#define OFF_T    4325376u  // T:   [b(16)][ky(32)][part(2)][i(64)][kx(64)]        4194304
// total ws: 8,519,680 floats = ~32.5 MB

// ---------------------------------------------------------------------------
// Kernel 0: build DFT bases (integer phase mod 256 for exactness).
// ---------------------------------------------------------------------------
__global__ __launch_bounds__(256) void k_bases(float* __restrict__ ws) {
  const int idx = blockIdx.x * 256 + threadIdx.x;
  if (idx >= 131072) return;
  const float TWO_PI_256 = 6.283185307179586f / 256.0f;
  const float s = 1.0f / 16.0f;  // ortho: 1/sqrt(256) per 1-D pass
  float val;
  if (idx < 16384) {               // forward W basis (B layout)
    int j = idx;
    int v = j & 1, l = (j >> 1) & 31, q = (j >> 6) & 63, kt = (j >> 12) & 1, ct = (j >> 13) & 1;
    int lo = l & 15, hi = l >> 4;
    int w = 4 * q + v + 2 * hi, ky = kt * 16 + lo;
    float th = TWO_PI_256 * (float)((w * ky) & 255);
    val = (ct == 0 ? cosf(th) : -sinf(th)) * s;          // e^{-i th}
  } else if (idx < 65536) {        // forward H basis (A layout), parts R,I,NI
    int j = idx - 16384;
    int v = j & 1, l = (j >> 1) & 31, q = (j >> 6) & 63, t = (j >> 12) & 3, p = j >> 14;
    int lo = l & 15, hi = l >> 4;
    int h = 4 * q + v + 2 * hi;
    int kxs = t * 16 + lo;
    int kx = (kxs < 32) ? kxs : (kxs + 192);             // bottom block: rows -32..-1
    float th = TWO_PI_256 * (float)((kx * h) & 255);
    val = (p == 0 ? cosf(th) : (p == 1 ? -sinf(th) : sinf(th))) * s;  // R, I, -I of e^{-i th}
  } else if (idx < 114688) {       // inverse H basis (A layout), parts R,I,NI
    int j = idx - 65536;
    int v = j & 1, l = (j >> 1) & 31, q = (j >> 6) & 15, t = (j >> 10) & 15, p = j >> 14;
    int lo = l & 15, hi = l >> 4;
    int h = t * 16 + lo;
    int k = 4 * q + v + 2 * hi;
    int kx = (k < 32) ? k : (k + 192);
    float th = TWO_PI_256 * (float)((kx * h) & 255);
    val = (p == 0 ? cosf(th) : (p == 1 ? sinf(th) : -sinf(th))) * s;  // R, I, -I of e^{+i th}
  } else {                         // inverse W basis (B layout), Hermitian-folded
    int j = idx - 114688;
    int v = j & 1, l = (j >> 1) & 31, q = (j >> 6) & 7, wt = (j >> 9) & 15, ct = j >> 13;
    int lo = l & 15, hi = l >> 4;
    int ky = 4 * q + v + 2 * hi;
    int w = wt * 16 + lo;
    float alpha = (ky == 0) ? 1.0f : 2.0f;
    float th = TWO_PI_256 * (float)((ky * w) & 255);
    val = (ct == 0 ? alpha * cosf(th) : -alpha * sinf(th)) * s;
  }
  ws[idx] = val;
}

// ---------------------------------------------------------------------------
// Kernel 1: forward truncated rDFT2, one block per (b, cin) channel.
//   F1: Y[h,ky] = sum_w x[h,w] e^{-i 2pi w ky/W}/16     (256x32 complex, LDS)
//       4 accumulators per wave: one A (x) load feeds 4 WMMAs -> x read once.
//   F2: X[kx,ky] = sum_h e^{-i 2pi kx h/H}/16 * Y[h,ky] (64 kept kx -> ws)
// ---------------------------------------------------------------------------
__global__ __launch_bounds__(256) void k_forward(const float* __restrict__ x,
                                                 float* __restrict__ ws) {
  __shared__ float Y[256 * 64];  // cols 0..31 = Re, 32..63 = Im  (64 KB)
  const int ch = blockIdx.x;     // b*64 + i
  const int tid = threadIdx.x, wv = tid >> 5, l = tid & 31, lo = l & 15, hi = l >> 4;
  const float* __restrict__ xc = x + (size_t)ch * 65536;
  const float* __restrict__ FW = ws + OFF_FW;

  // ---- F1: M=h (16 tiles, 2 per wave), N = {Re,Im}x{kyt0,kyt1}, K=256 ----
  for (int tm = wv; tm < 16; tm += 8) {
    const float* ap = xc + (tm * 16 + lo) * 256 + 2 * hi;
    const float* bp = FW + 2 * l;
    if (tm < 8)  // prefetch next row-tile of x (global_prefetch_b8)
      __builtin_prefetch(xc + ((tm + 8) * 16 + lo) * 256, 0, 0);
    v8f a0 = zero8(), a1 = zero8(), a2 = zero8(), a3 = zero8();
#pragma unroll 2
    for (int q = 0; q < 64; ++q) {
      v2f a  = *(const v2f*)(ap + 4 * q);
      v2f b0 = *(const v2f*)(bp + ((0 * 64 + q) << 6));
      v2f b1 = *(const v2f*)(bp + ((1 * 64 + q) << 6));
      v2f b2 = *(const v2f*)(bp + ((2 * 64 + q) << 6));
      v2f b3 = *(const v2f*)(bp + ((3 * 64 + q) << 6));
      a0 = wmma4(a, b0, a0);
      a1 = wmma4(a, b1, a1);
      a2 = wmma4(a, b2, a2);
      a3 = wmma4(a, b3, a3);
    }
    int r0 = tm * 16 + 8 * hi;
#pragma unroll
    for (int r = 0; r < 8; ++r) {
      Y[(r0 + r) * 64 +  0 + lo] = a0[r];  // Re, ky 0..15
      Y[(r0 + r) * 64 + 16 + lo] = a1[r];  // Re, ky 16..31
      Y[(r0 + r) * 64 + 32 + lo] = a2[r];  // Im, ky 0..15
      Y[(r0 + r) * 64 + 48 + lo] = a3[r];  // Im, ky 16..31
    }
  }
  __syncthreads();

  // ---- F2: complex GEMM, A = packed e^{-i th} (parts R/I/NI), B = Y ----
  const float* __restrict__ FH = ws + OFF_FH;
  float* __restrict__ XFT = ws + OFF_XFT;
  const int b_ = ch >> 6, ci = ch & 63;
  for (int t = wv; t < 16; t += 8) {
    int tmx = t >> 2, tn = t & 3, pr = tn >> 1, kt = tn & 1;
    int pA1 = 0, cb1, pA2, cb2;
    if (pr == 0) { cb1 = 0;  pA2 = 2; cb2 = 32; }   // XR = R*YR + (-I)*YI
    else         { cb1 = 32; pA2 = 1; cb2 = 0;  }   // XI = R*YI +   I *YR
    v8f acc = zero8();
#pragma unroll 4
    for (int q = 0; q < 64; ++q) {
      v2f a = *(const v2f*)(FH + (((pA1 * 4 + tmx) * 64 + q) << 6) + 2 * l);
      int k = 4 * q + 2 * hi;
      v2f b = { Y[k * 64 + cb1 + kt * 16 + lo], Y[(k + 1) * 64 + cb1 + kt * 16 + lo] };
      acc = wmma4(a, b, acc);
    }
#pragma unroll 4
    for (int q = 0; q < 64; ++q) {
      v2f a = *(const v2f*)(FH + (((pA2 * 4 + tmx) * 64 + q) << 6) + 2 * l);
      int k = 4 * q + 2 * hi;
      v2f b = { Y[k * 64 + cb2 + kt * 16 + lo], Y[(k + 1) * 64 + cb2 + kt * 16 + lo] };
      acc = wmma4(a, b, acc);
    }
    int ky = kt * 16 + lo;
#pragma unroll
    for (int r = 0; r < 8; ++r) {
      int kx = tmx * 16 + 8 * hi + r;
      XFT[((((size_t)b_ * 64 + kx) * 2 + pr) * 64 + ci) * 32 + ky] = acc[r];
    }
  }
}

// ---------------------------------------------------------------------------
// Kernel 2: einsum 'bixy,iox->boxy'. One block per kx (64), loop over b.
//   T[ky,o] = sum_i X^T[ky,i] * W[i,o]  (complex, K=64)
// ---------------------------------------------------------------------------
__global__ __launch_bounds__(256) void k_mix1(const float* __restrict__ w1,
                                              const float* __restrict__ w2,
                                              float* __restrict__ ws) {
  __shared__ float WR[4096], WI[4096], WNI[4096];  // 48 KB
  const int kx = blockIdx.x;
  const int tid = threadIdx.x, wv = tid >> 5, l = tid & 31, lo = l & 15, hi = l >> 4;
  const float* __restrict__ wsrc = (kx < 32) ? w1 : w2;
  const int kxw = kx & 31;
  for (int idx = tid; idx < 4096; idx += 256) {     // idx = i*64 + o
    const float* p = wsrc + ((size_t)idx * 32 + kxw) * 2;
    float re = p[0], im = p[1];
    WR[idx] = re; WI[idx] = im; WNI[idx] = -im;
  }
  __syncthreads();
  const float* __restrict__ XFT = ws + OFF_XFT;
  float* __restrict__ T = ws + OFF_T;
  for (int b_ = 0; b_ < 16; ++b_) {
    const float* Xb = XFT + ((size_t)b_ * 64 + kx) * 2 * 2048;  // part stride 64*32
    for (int t = wv; t < 16; t += 8) {
      int pr = t >> 3, rem = t & 7, tmy = rem >> 2, tno = rem & 3;
      const float *B1, *B2;
      if (pr == 0) { B1 = WR; B2 = WNI; }            // TR = XR*WR + XI*(-WI)
      else         { B1 = WI; B2 = WR;  }            // TI = XR*WI + XI*WR
      v8f acc = zero8();
#pragma unroll 4
      for (int q = 0; q < 16; ++q) {
        int k = 4 * q + 2 * hi, m = tmy * 16 + lo, n = tno * 16 + lo;
        v2f a = { Xb[0 * 2048 + k * 32 + m], Xb[0 * 2048 + (k + 1) * 32 + m] };
        v2f b = { B1[k * 64 + n], B1[(k + 1) * 64 + n] };
        acc = wmma4(a, b, acc);
      }
#pragma unroll 4
      for (int q = 0; q < 16; ++q) {
        int k = 4 * q + 2 * hi, m = tmy * 16 + lo, n = tno * 16 + lo;
        v2f a = { Xb[1 * 2048 + k * 32 + m], Xb[1 * 2048 + (k + 1) * 32 + m] };
        v2f b = { B2[k * 64 + n], B2[(k + 1) * 64 + n] };
        acc = wmma4(a, b, acc);
      }
      int o = tno * 16 + lo;
#pragma unroll
      for (int r = 0; r < 8; ++r) {
        int ky = tmy * 16 + 8 * hi + r;
        T[((((size_t)b_ * 32 + ky) * 2 + pr) * 64 + o) * 64 + kx] = acc[r];
      }
    }
  }
}

// ---------------------------------------------------------------------------
// Kernel 3: einsum 'bixy,ioy->boxy'. One block per (b, ky) (512 blocks).
//   Y[kx,o] = sum_i T[kx,i] * Wy[i,o]  (complex, K=64)
// ---------------------------------------------------------------------------
__global__ __launch_bounds__(256) void k_mix2(const float* __restrict__ wy,
                                              float* __restrict__ ws) {
  __shared__ float WR[4096], WI[4096], WNI[4096];
  const int blk = blockIdx.x, b_ = blk >> 5, ky = blk & 31;
  const int tid = threadIdx.x, wv = tid >> 5, l = tid & 31, lo = l & 15, hi = l >> 4;
  for (int idx = tid; idx < 4096; idx += 256) {
    const float* p = wy + ((size_t)idx * 32 + ky) * 2;
    float re = p[0], im = p[1];
    WR[idx] = re; WI[idx] = im; WNI[idx] = -im;
  }
  __syncthreads();
  const float* __restrict__ Tb = ws + OFF_T + ((size_t)b_ * 32 + ky) * 2 * 4096; // part stride 64*64
  float* __restrict__ YFT = ws + OFF_YFT;
  for (int t = wv; t < 32; t += 8) {
    int pr = t >> 4, rem = t & 15, tmx = rem >> 2, tno = rem & 3;
    const float *B1, *B2;
    if (pr == 0) { B1 = WR; B2 = WNI; } else { B1 = WI; B2 = WR; }
    v8f acc = zero8();
#pragma unroll 4
    for (int q = 0; q < 16; ++q) {
      int k = 4 * q + 2 * hi, m = tmx * 16 + lo, n = tno * 16 + lo;
      v2f a = { Tb[0 * 4096 + k * 64 + m], Tb[0 * 4096 + (k + 1) * 64 + m] };
      v2f b = { B1[k * 64 + n], B1[(k + 1) * 64 + n] };
      acc = wmma4(a, b, acc);
    }
#pragma unroll 4
    for (int q = 0; q < 16; ++q) {
      int k = 4 * q + 2 * hi, m = tmx * 16 + lo, n = tno * 16 + lo;
      v2f a = { Tb[1 * 4096 + k * 64 + m], Tb[1 * 4096 + (k + 1) * 64 + m] };
      v2f b = { B2[k * 64 + n], B2[(k + 1) * 64 + n] };
      acc = wmma4(a, b, acc);
    }
    int o = tno * 16 + lo;
#pragma unroll
    for (int r = 0; r < 8; ++r) {
      int kx = tmx * 16 + 8 * hi + r;
      YFT[(((size_t)b_ * 64 + o) * 2 + pr) * 2048 + kx * 32 + ky] = acc[r];
    }
  }
}

// ---------------------------------------------------------------------------
// Kernel 4: inverse truncated rDFT2, one block per (b, cout) channel.
//   IA: V[h,ky] = sum_{kx in S} e^{+i 2pi kx h/H}/16 * Yft[kx,ky]  (LDS)
//   IB: out[h,w] = sum_ky VR*alpha*cos/16 - VI*alpha*sin/16        (global)
//       4 accumulators per wave: one A (LDS) load feeds 4 WMMAs.
// ---------------------------------------------------------------------------
__global__ __launch_bounds__(256) void k_inverse(const float* __restrict__ ws,
                                                 float* __restrict__ out) {
  __shared__ float V[256 * 64];  // cols 0..31 Re, 32..63 Im
  const int ch = blockIdx.x;     // b*64 + o
  const int tid = threadIdx.x, wv = tid >> 5, l = tid & 31, lo = l & 15, hi = l >> 4;
  const float* __restrict__ IH = ws + OFF_IH;
  const float* __restrict__ Yc = ws + OFF_YFT + (size_t)ch * 2 * 2048;

  for (int t = wv; t < 64; t += 8) {
    int tmh = t >> 2, tn = t & 3, pr = tn >> 1, kt = tn & 1;
    int pA1 = 0, pB1, pA2, pB2;
    if (pr == 0) { pB1 = 0; pA2 = 2; pB2 = 1; }   // VR = R*XR + (-I)*XI
    else         { pB1 = 1; pA2 = 1; pB2 = 0; }   // VI = R*XI +   I *XR
    v8f acc = zero8();
#pragma unroll 4
    for (int q = 0; q < 16; ++q) {
      v2f a = *(const v2f*)(IH + (((pA1 * 16 + tmh) * 16 + q) << 6) + 2 * l);
      int k = 4 * q + 2 * hi;
      v2f b = { Yc[pB1 * 2048 + k * 32 + kt * 16 + lo],
                Yc[pB1 * 2048 + (k + 1) * 32 + kt * 16 + lo] };
      acc = wmma4(a, b, acc);
    }
#pragma unroll 4
    for (int q = 0; q < 16; ++q) {
      v2f a = *(const v2f*)(IH + (((pA2 * 16 + tmh) * 16 + q) << 6) + 2 * l);
      int k = 4 * q + 2 * hi;
      v2f b = { Yc[pB2 * 2048 + k * 32 + kt * 16 + lo],
                Yc[pB2 * 2048 + (k + 1) * 32 + kt * 16 + lo] };
      acc = wmma4(a, b, acc);
    }
    int col = pr * 32 + kt * 16 + lo, r0 = tmh * 16 + 8 * hi;
#pragma unroll
    for (int r = 0; r < 8; ++r) V[(r0 + r) * 64 + col] = acc[r];
  }
  __syncthreads();

  // ---- IB: 16 h-tiles (2/wave) x 16 w-tiles (groups of 4, 4 accs) ----
  const float* __restrict__ IW = ws + OFF_IW;
  float* __restrict__ oc = out + (size_t)ch * 65536;
  for (int tm = wv; tm < 16; tm += 8) {
    const float* ar = &V[(tm * 16 + lo) * 64];
    for (int twg = 0; twg < 4; ++twg) {
      v8f a0 = zero8(), a1 = zero8(), a2 = zero8(), a3 = zero8();
#pragma unroll
      for (int ct = 0; ct < 2; ++ct) {   // ct=0: VR*cos-table, ct=1: VI*sin-table
#pragma unroll
        for (int q = 0; q < 8; ++q) {
          v2f a  = *(const v2f*)(ar + ct * 32 + 4 * q + 2 * hi);
          v2f b0 = *(const v2f*)(IW + (((ct * 16 + twg * 4 + 0) * 8 + q) << 6) + 2 * l);
          v2f b1 = *(const v2f*)(IW + (((ct * 16 + twg * 4 + 1) * 8 + q) << 6) + 2 * l);
          v2f b2 = *(const v2f*)(IW + (((ct * 16 + twg * 4 + 2) * 8 + q) << 6) + 2 * l);
          v2f b3 = *(const v2f*)(IW + (((ct * 16 + twg * 4 + 3) * 8 + q) << 6) + 2 * l);
          a0 = wmma4(a, b0, a0);
          a1 = wmma4(a, b1, a1);
          a2 = wmma4(a, b2, a2);
          a3 = wmma4(a, b3, a3);
        }
      }
#pragma unroll
      for (int r = 0; r < 8; ++r) {
        int row = tm * 16 + 8 * hi + r;
        oc[row * 256 + (twg * 4 + 0) * 16 + lo] = a0[r];
        oc[row * 256 + (twg * 4 + 1) * 16 + lo] = a1[r];
        oc[row * 256 + (twg * 4 + 2) * 16 + lo] = a2[r];
        oc[row * 256 + (twg * 4 + 3) * 16 + lo] = a3[r];
      }
    }
  }
}

// ---------------------------------------------------------------------------
extern "C" void kernel_launch(void* const* d_in, const int* in_sizes, int n_in,
                              void* d_out, int out_size, void* d_ws, size_t ws_size,
                              hipStream_t stream) {
  (void)in_sizes; (void)n_in; (void)out_size; (void)ws_size;
  const float* x  = (const float*)d_in[0];
  const float* w1 = (const float*)d_in[1];
  const float* w2 = (const float*)d_in[2];
  const float* wy = (const float*)d_in[3];
  float* ws  = (float*)d_ws;
  float* out = (float*)d_out;

  k_bases  <<<512,  256, 0, stream>>>(ws);
  k_forward<<<1024, 256, 0, stream>>>(x, ws);
  k_mix1   <<<64,   256, 0, stream>>>(w1, w2, ws);
  k_mix2   <<<512,  256, 0, stream>>>(wy, ws);
  k_inverse<<<1024, 256, 0, stream>>>(ws, out);
}